// Model_56788057588352
// MI455X (gfx1250) — compile-verified
//
#include <hip/hip_runtime.h>
#include <hip/hip_bf16.h>

// ---------------- problem constants ----------------
#define Nn   32
#define Cc   64
#define Tt   256
#define Vv   25
#define OUTC 64
#define INTER 16
#define TV   (Tt*Vv)            // 6400
#define CTV  (Cc*TV)            // 409600
#define PLANE ((size_t)Nn*CTV)  // 13,107,200 elements per real/imag tensor

typedef float v2f __attribute__((ext_vector_type(2)));
typedef float v8f __attribute__((ext_vector_type(8)));

// ---------------------------------------------------------------------------
// Fused Wa+Wb complex 1x1 conv as f32 WMMA GEMM (stages x once):
//   a1[o][pos] = ba[o] + sum_k Wa[o][k]*in[k][pos], same for a2/Wb
//   k<64 -> x_r channel k, k>=64 -> x_i channel k-64; rows o<16 real, o>=16 imag
// Block: 256 thr (8 waves). Grid: (32 pos-blocks of 200, Nn).
// LDS: 128 K-rows x 208 cols (200 real + 8 zero pad) = 104 KB.
// Row-tiles: rt4 = 0,1 -> Wa ; rt4 = 2,3 -> Wb  (rt4 = wave%4, wave-uniform).
// ---------------------------------------------------------------------------
__global__ void conv_ab_wmma(const float* __restrict__ x_r,
                             const float* __restrict__ x_i,
                             const float* __restrict__ Wa,   // [32,128]
                             const float* __restrict__ ba,   // [32]
                             const float* __restrict__ Wb,   // [32,128]
                             const float* __restrict__ bb,   // [32]
                             float* __restrict__ a1r, float* __restrict__ a1i,
                             float* __restrict__ a2r, float* __restrict__ a2i) {
  extern __shared__ float lds[];                    // 128*208 floats
  const int n = blockIdx.y;
  const int posBase = blockIdx.x * 200;
  const int tid = threadIdx.x;

  // stage input tile into LDS (columns 200..207 zero)
  for (int idx = tid; idx < 128 * 208; idx += 256) {
    int k = idx / 208;
    int p = idx - k * 208;
    float v = 0.0f;
    if (p < 200) {
      const float* src = (k < Cc) ? x_r : x_i;
      v = src[(size_t)n * CTV + (size_t)(k & (Cc - 1)) * TV + posBase + p];
    }
    lds[idx] = v;
  }
  __syncthreads();

  const int wave = tid >> 5;
  const int lane = tid & 31;
  const int rt4 = wave % 4;            // wave-uniform (8 % 4 == 0)
  const int rt = rt4 & 1;              // row tile inside the 32-row weight
  const int lm = lane & 15;
  const int khalf = (lane >> 4) * 2;   // 0 or 2

  const float* Wsel = (rt4 < 2) ? Wa : Wb;
  const float* bsel = (rt4 < 2) ? ba : bb;
  float* outR = (rt4 < 2) ? a1r : a2r;
  float* outI = (rt4 < 2) ? a1i : a2i;

  // preload all 32 A-fragments (weights) for this wave's 16-row tile
  v2f afrag[32];
  const float* Wrow = Wsel + (size_t)(rt * 16 + lm) * 128;
#pragma unroll
  for (int ks = 0; ks < 32; ++ks) {
    afrag[ks].x = Wrow[ks * 4 + khalf];
    afrag[ks].y = Wrow[ks * 4 + khalf + 1];
  }

  for (int task = wave; task < 52; task += 8) {     // 4 row-tiles x 13 col-tiles
    const int ct = task / 4;
    const int col = ct * 16 + lm;
    v8f acc = {0.f, 0.f, 0.f, 0.f, 0.f, 0.f, 0.f, 0.f};
#pragma unroll
    for (int ks = 0; ks < 32; ++ks) {
      const int kr = ks * 4 + khalf;
      v2f b;
      b.x = lds[kr * 208 + col];
      b.y = lds[(kr + 1) * 208 + col];
      acc = __builtin_amdgcn_wmma_f32_16x16x4_f32(
          false, afrag[ks], false, b, (short)0, acc, false, false);
    }
    if (col < 200) {
      const int gp = posBase + col;
      const int mbase = rt * 16 + (lane >> 4) * 8;  // row in [0,32)
#pragma unroll
      for (int r = 0; r < 8; ++r) {
        const int o = mbase + r;
        const float val = acc[r] + bsel[o];
        float* dst = (o < INTER) ? outR : outI;
        const int oo = (o < INTER) ? o : o - INTER;
        dst[(size_t)n * INTER * TV + (size_t)oo * TV + gp] = val;
      }
    }
  }
}

// ---------------------------------------------------------------------------
// Fused (x @ S) -> Wd complex 1x1 conv, all in LDS (no z round-trip):
// A 200-position window is exactly 8 full t-rows, so the block stages
// x[64ch][8t][25v] (re+im) + S, builds the 128x208 z tile in LDS, then runs
// the f32 WMMA GEMM  y[o][pos] (+)= bd[o] + sum_k Wd[o][k] * z[k][pos].
// LDS: 26624 (z) + 2*12800 (x) + 2*625 (S) floats = ~209 KB.
// ---------------------------------------------------------------------------
__global__ void conv_wd_fused(const float* __restrict__ x_r,
                              const float* __restrict__ x_i,
                              const float* __restrict__ Sr,
                              const float* __restrict__ Si,
                              const float* __restrict__ W,    // [128,128]
                              const float* __restrict__ bias, // [128]
                              float* __restrict__ yr,
                              float* __restrict__ yi,
                              int accum) {
  extern __shared__ float lds[];
  float* zb  = lds;                 // 128*208 = 26624
  float* xbr = lds + 26624;         // 64*200  = 12800
  float* xbi = xbr + 12800;         // 12800
  float* sbr = xbi + 12800;         // 625
  float* sbi = sbr + 625;           // 625

  const int n = blockIdx.y;
  const int posBase = blockIdx.x * 200;    // = (8*blockIdx.x)*25, 8 whole t-rows
  const int tid = threadIdx.x;

  // stage x tile (contiguous per channel) and S
  for (int idx = tid; idx < 64 * 200; idx += 256) {
    const int c = idx / 200;
    const int p = idx - c * 200;
    const size_t g = (size_t)n * CTV + (size_t)c * TV + posBase + p;
    xbr[idx] = x_r[g];
    xbi[idx] = x_i[g];
  }
  for (int j = tid; j < 625; j += 256) {
    sbr[j] = Sr[n * 625 + j];
    sbi[j] = Si[n * 625 + j];
  }
  __syncthreads();

  // build z tile in LDS: z[k][p], k<64 real-part channels, k>=64 imag
  for (int idx = tid; idx < 128 * 208; idx += 256) {
    const int k = idx / 208;
    const int p = idx - k * 208;
    float val = 0.0f;
    if (p < 200) {
      const int c = k & (Cc - 1);
      const int lt = p / 25;
      const int w = p - lt * 25;
      const float* xr0 = xbr + c * 200 + lt * 25;
      const float* xi0 = xbi + c * 200 + lt * 25;
      if (k < Cc) {
#pragma unroll
        for (int v = 0; v < 25; ++v)
          val += xr0[v] * sbr[v * 25 + w] - xi0[v] * sbi[v * 25 + w];
      } else {
#pragma unroll
        for (int v = 0; v < 25; ++v)
          val += xr0[v] * sbi[v * 25 + w] + xi0[v] * sbr[v * 25 + w];
      }
    }
    zb[idx] = val;
  }
  __syncthreads();

  // WMMA GEMM: 8 row-tiles (M=128) x 13 col-tiles, K=128 in 32 steps of 4
  const int wave = tid >> 5;
  const int lane = tid & 31;
  const int rt = wave;                 // RT == 8 == #waves
  const int lm = lane & 15;
  const int khalf = (lane >> 4) * 2;

  v2f afrag[32];
  const float* Wrow = W + (size_t)(rt * 16 + lm) * 128;
#pragma unroll
  for (int ks = 0; ks < 32; ++ks) {
    afrag[ks].x = Wrow[ks * 4 + khalf];
    afrag[ks].y = Wrow[ks * 4 + khalf + 1];
  }

  for (int ct = 0; ct < 13; ++ct) {
    const int col = ct * 16 + lm;
    v8f acc = {0.f, 0.f, 0.f, 0.f, 0.f, 0.f, 0.f, 0.f};
#pragma unroll
    for (int ks = 0; ks < 32; ++ks) {
      const int kr = ks * 4 + khalf;
      v2f b;
      b.x = zb[kr * 208 + col];
      b.y = zb[(kr + 1) * 208 + col];
      acc = __builtin_amdgcn_wmma_f32_16x16x4_f32(
          false, afrag[ks], false, b, (short)0, acc, false, false);
    }
    if (col < 200) {
      const int gp = posBase + col;
      const int mbase = rt * 16 + (lane >> 4) * 8;
#pragma unroll
      for (int r = 0; r < 8; ++r) {
        const int o = mbase + r;
        const float val = acc[r] + bias[o];
        float* dst = (o < OUTC) ? yr : yi;
        const int oo = (o < OUTC) ? o : o - OUTC;
        const size_t off = (size_t)n * CTV + (size_t)oo * TV + gp;
        if (accum) dst[off] += val; else dst[off] = val;
      }
    }
  }
}

// ---------------------------------------------------------------------------
// S[n,v,w] = (1/(INTER*T)) * sum_{kk<4096} A1c[n,kk,v] * A2c[n,kk,w]  (complex)
// ---------------------------------------------------------------------------
__global__ void compute_s_kernel(const float* __restrict__ a1r,
                                 const float* __restrict__ a1i,
                                 const float* __restrict__ a2r,
                                 const float* __restrict__ a2i,
                                 float* __restrict__ Sr,
                                 float* __restrict__ Si) {
  __shared__ float s1r[3200], s1i[3200], s2r[3200], s2i[3200];
  const int n = blockIdx.x;
  const int tid = threadIdx.x;
  float accR[3] = {0.f, 0.f, 0.f};
  float accI[3] = {0.f, 0.f, 0.f};
  const size_t base = (size_t)n * (INTER * TV);

  for (int kk0 = 0; kk0 < 4096; kk0 += 128) {
    const size_t cb = base + (size_t)kk0 * Vv;
    for (int j = tid; j < 3200; j += 256) {
      s1r[j] = a1r[cb + j]; s1i[j] = a1i[cb + j];
      s2r[j] = a2r[cb + j]; s2i[j] = a2i[cb + j];
    }
    __syncthreads();
    for (int pp = 0; pp < 3; ++pp) {
      const int pair = tid + pp * 256;
      if (pair < 625) {
        const int v = pair / 25;
        const int w = pair - v * 25;
        float aR = accR[pp], aI = accI[pp];
        for (int k = 0; k < 128; ++k) {
          const float ar = s1r[k * 25 + v], ai = s1i[k * 25 + v];
          const float br = s2r[k * 25 + w], bi = s2i[k * 25 + w];
          aR += ar * br - ai * bi;
          aI += ar * bi + ai * br;
        }
        accR[pp] = aR; accI[pp] = aI;
      }
    }
    __syncthreads();
  }
  const float sc = 1.0f / 4096.0f;
  for (int pp = 0; pp < 3; ++pp) {
    const int pair = tid + pp * 256;
    if (pair < 625) {
      Sr[n * 625 + pair] = accR[pp] * sc;
      Si[n * 625 + pair] = accI[pp] * sc;
    }
  }
}

// ---------------------------------------------------------------------------
// Complex softmax over rows (dim v) per column w, then += (A + PA).
// ---------------------------------------------------------------------------
__global__ void softmax25_kernel(float* __restrict__ Sr, float* __restrict__ Si,
                                 const float* __restrict__ Ai,
                                 const float* __restrict__ PAi) {
  const int n = blockIdx.x;
  const int w = threadIdx.x;
  if (w >= 25) return;
  float er[25], ei[25];
  float sumr = 0.f, sumi = 0.f;
  for (int v = 0; v < 25; ++v) {
    const float a = Sr[n * 625 + v * 25 + w];
    const float b = Si[n * 625 + v * 25 + w];
    const float e = expf(a);
    const float cr = e * cosf(b);
    const float ci = e * sinf(b);
    er[v] = cr; ei[v] = ci;
    sumr += cr; sumi += ci;
  }
  const float inv = 1.0f / (sumr * sumr + sumi * sumi);
  for (int v = 0; v < 25; ++v) {
    const float rr = (er[v] * sumr + ei[v] * sumi) * inv;
    const float ri = (ei[v] * sumr - er[v] * sumi) * inv;
    Sr[n * 625 + v * 25 + w] = rr + Ai[v * 25 + w] + PAi[v * 25 + w];
    Si[n * 625 + v * 25 + w] = ri;
  }
}

// ---------------------------------------------------------------------------
// Per-channel stats: stats[o] = {sum re, sum im, sum |y|^2}. One block per o.
// ---------------------------------------------------------------------------
__global__ void bn_stats_kernel(const float* __restrict__ yr,
                                const float* __restrict__ yi,
                                float* __restrict__ stats) {
  __shared__ float rs[256], ms[256], qs[256];
  const int o = blockIdx.x;
  const int tid = threadIdx.x;
  float sr = 0.f, si = 0.f, sq = 0.f;
  for (int nn = 0; nn < Nn; ++nn) {
    const size_t base = (size_t)nn * CTV + (size_t)o * TV;
    for (int p = tid; p < TV; p += 256) {
      const float a = yr[base + p], b = yi[base + p];
      sr += a; si += b; sq += a * a + b * b;
    }
  }
  rs[tid] = sr; ms[tid] = si; qs[tid] = sq;
  __syncthreads();
  for (int s = 128; s > 0; s >>= 1) {
    if (tid < s) { rs[tid] += rs[tid + s]; ms[tid] += ms[tid + s]; qs[tid] += qs[tid + s]; }
    __syncthreads();
  }
  if (tid == 0) {
    stats[o * 3 + 0] = rs[0];
    stats[o * 3 + 1] = ms[0];
    stats[o * 3 + 2] = qs[0];
  }
}

// ---------------------------------------------------------------------------
// yn = (y - mu)/sqrt(var+eps)*w + b + x ; write [2,N,OUT,T,V]
// ---------------------------------------------------------------------------
__global__ void finalize_kernel(const float* __restrict__ yr,
                                const float* __restrict__ yi,
                                const float* __restrict__ xr,
                                const float* __restrict__ xi,
                                const float* __restrict__ stats,
                                const float* __restrict__ bnw,
                                const float* __restrict__ bnb,
                                float* __restrict__ out) {
  const size_t idx = (size_t)blockIdx.x * 256 + threadIdx.x;   // < PLANE
  const int o = (int)((idx % CTV) / TV);
  const float cnt = (float)Nn * (float)TV;                      // 204800
  const float mur = stats[o * 3 + 0] / cnt;
  const float mui = stats[o * 3 + 1] / cnt;
  float var = stats[o * 3 + 2] / cnt - (mur * mur + mui * mui);
  const float s = bnw[o] * rsqrtf(var + 1e-5f);
  const float orv = (yr[idx] - mur) * s + bnb[o] + xr[idx];
  const float oiv = (yi[idx] - mui) * s + xi[idx];
  out[idx] = orv;
  out[PLANE + idx] = oiv;
}

// ---------------------------------------------------------------------------
extern "C" void kernel_launch(void* const* d_in, const int* in_sizes, int n_in,
                              void* d_out, int out_size, void* d_ws, size_t ws_size,
                              hipStream_t stream) {
  (void)in_sizes; (void)n_in; (void)out_size; (void)ws_size;
  const float* x_r = (const float*)d_in[0];
  const float* x_i = (const float*)d_in[1];
  const float* A   = (const float*)d_in[2];
  const float* PA  = (const float*)d_in[3];
  const float* Wa  = (const float*)d_in[4];
  const float* ba  = (const float*)d_in[5];
  const float* Wb  = (const float*)d_in[6];
  const float* bb  = (const float*)d_in[7];
  const float* Wd  = (const float*)d_in[8];
  const float* bd  = (const float*)d_in[9];
  const float* bnw = (const float*)d_in[10];
  const float* bnb = (const float*)d_in[11];
  float* out = (float*)d_out;

  float* w = (float*)d_ws;
  float* a1r = w;                          // N*16*TV = 3,276,800 each
  float* a1i = w + 3276800;
  float* a2r = w + 6553600;
  float* a2i = w + 9830400;
  float* yr  = w + PLANE;                  // PLANE (a-bufs total exactly PLANE)
  float* yi  = w + 2 * PLANE;              // PLANE
  float* Sr  = w + 3 * PLANE;              // N*625 = 20,000
  float* Si  = Sr + 20000;
  float* stats = Si + 20000;               // 192

  const size_t ldsAB = (size_t)128 * 208 * sizeof(float);            // 104 KB
  const size_t ldsWD = (size_t)(26624 + 2 * 12800 + 2 * 625) * sizeof(float); // ~209 KB

  for (int i = 0; i < 3; ++i) {
    conv_ab_wmma<<<dim3(32, Nn), 256, ldsAB, stream>>>(
        x_r, x_i,
        Wa + (size_t)i * 32 * 128, ba + i * 32,
        Wb + (size_t)i * 32 * 128, bb + i * 32,
        a1r, a1i, a2r, a2i);
    compute_s_kernel<<<Nn, 256, 0, stream>>>(a1r, a1i, a2r, a2i, Sr, Si);
    softmax25_kernel<<<Nn, 32, 0, stream>>>(Sr, Si, A + (size_t)i * 625, PA + (size_t)i * 625);
    conv_wd_fused<<<dim3(32, Nn), 256, ldsWD, stream>>>(
        x_r, x_i, Sr, Si,
        Wd + (size_t)i * 128 * 128, bd + i * 128,
        yr, yi, (i > 0) ? 1 : 0);
  }
  bn_stats_kernel<<<OUTC, 256, 0, stream>>>(yr, yi, stats);
  finalize_kernel<<<(int)(PLANE / 256), 256, 0, stream>>>(
      yr, yi, x_r, x_i, stats, bnw, bnb, out);
}